// Pifold_featurizer_28845000360670
// MI455X (gfx1250) — compile-verified
//
#include <hip/hip_runtime.h>
#include <math.h>

// Problem constants (from setup_inputs): B=8, N=1024, top_k=30, num_rbf=16
#define BB 8
#define NN 1024
#define KK 30
#define NRBF 16
#define VFEAT 213   // 12 pairs * 16 rbf + 12 angles + 9 direct
#define EFEAT 416   // 25 pairs * 16 rbf + 4 quat + 12 direct

typedef float v2f __attribute__((ext_vector_type(2)));
typedef float v8f __attribute__((ext_vector_type(8)));

// Pair tables (atom ids: 0=N 1=Ca 2=C 3=O 4..6=virtual)
__constant__ int c_NPa[12] = {1,1,1,0,0,3, 5,4,6,4,6,5};
__constant__ int c_NPb[12] = {0,2,3,2,3,2, 4,5,4,6,5,6};
__constant__ int c_EPa[25] = {1,1,2,1,0,1,3,2,2,0,2,3,0,0,3,3, 4,5,5,4,6,6,4,6,5};
__constant__ int c_EPb[25] = {1,2,1,0,1,3,1,2,0,2,3,2,0,3,0,3, 4,5,4,5,6,4,6,5,6};

struct F3 { float x, y, z; };
__device__ __forceinline__ F3 mk3(float x, float y, float z){ F3 r; r.x=x; r.y=y; r.z=z; return r; }
__device__ __forceinline__ F3 add3(F3 a, F3 b){ return mk3(a.x+b.x, a.y+b.y, a.z+b.z); }
__device__ __forceinline__ F3 sub3(F3 a, F3 b){ return mk3(a.x-b.x, a.y-b.y, a.z-b.z); }
__device__ __forceinline__ F3 scl3(F3 a, float s){ return mk3(a.x*s, a.y*s, a.z*s); }
__device__ __forceinline__ float dot3(F3 a, F3 b){ return a.x*b.x + a.y*b.y + a.z*b.z; }
__device__ __forceinline__ F3 cross3(F3 a, F3 b){
  return mk3(a.y*b.z - a.z*b.y, a.z*b.x - a.x*b.z, a.x*b.y - a.y*b.x);
}
__device__ __forceinline__ F3 normeps(F3 a){              // reference _normalize: v / max(|v|, 1e-12)
  float n = fmaxf(sqrtf(dot3(a,a)), 1e-12f);
  return scl3(a, 1.0f/n);
}
__device__ __forceinline__ float comp3(F3 a, int i){ return i==0 ? a.x : (i==1 ? a.y : a.z); }
__device__ __forceinline__ float signf(float x){ return (x>0.f)?1.f:((x<0.f)?-1.f:0.f); }
__device__ __forceinline__ float clampc(float x){ return fminf(fmaxf(x, -1.f+1e-7f), 1.f-1e-7f); }
__device__ __forceinline__ float rbfv(float D, int t){    // mu = linspace(2,22,16), sigma = 1.25
  float mu = 2.0f + (4.0f/3.0f)*(float)t;
  float u  = (D - mu) * 0.8f;
  return expf(-u*u);
}

__device__ __forceinline__ F3 loadAtom(const float* __restrict__ X, int b, int n, int a){
  const float* p = X + (((size_t)b*NN + n)*4 + a)*3;
  return mk3(p[0], p[1], p[2]);
}
// Backbone chain: Xb = X[:,:,:3,:].reshape(B,3N,3); position p -> residue p/3, atom p%3
__device__ __forceinline__ F3 chainAtom(const float* __restrict__ X, int b, int p){
  const float* q = X + (((size_t)b*NN + (p/3))*4 + (p%3))*3;
  return mk3(q[0], q[1], q[2]);
}
__device__ __forceinline__ F3 chainU(const float* __restrict__ X, int b, int i){
  return normeps(sub3(chainAtom(X, b, i+1), chainAtom(X, b, i)));
}
// Qm(n): rows [b1, n0, b1 x n0] from u0=U(3n), u1=U(3n+1); zero for n == N-1 (padded row)
__device__ __forceinline__ void computeQ(const float* __restrict__ X, int b, int n, F3 Q[3]){
  if (n >= NN-1) { Q[0]=mk3(0,0,0); Q[1]=mk3(0,0,0); Q[2]=mk3(0,0,0); return; }
  F3 u0 = chainU(X, b, 3*n);
  F3 u1 = chainU(X, b, 3*n+1);
  F3 b1 = normeps(sub3(u0, u1));
  F3 n0 = normeps(cross3(u0, u1));
  Q[0]=b1; Q[1]=n0; Q[2]=cross3(b1, n0);
}
// atoms: 0=N 1=Ca 2=C 3=O 4..6 = virtual atoms (constant-index use only)
__device__ __forceinline__ void buildAtoms(const float* __restrict__ X,
                                           const float* __restrict__ vraw,
                                           int b, int n, F3 out[7]){
  F3 aN = loadAtom(X,b,n,0), aCa = loadAtom(X,b,n,1), aC = loadAtom(X,b,n,2), aO = loadAtom(X,b,n,3);
  F3 vb = sub3(aCa, aN);
  F3 vc = sub3(aC, aCa);
  F3 va = cross3(vb, vc);
  out[0]=aN; out[1]=aCa; out[2]=aC; out[3]=aO;
  #pragma unroll
  for (int i=0;i<3;++i){
    F3 r = mk3(vraw[3*i], vraw[3*i+1], vraw[3*i+2]);
    float inv = 1.0f / sqrtf(dot3(r,r));
    r = scl3(r, inv);
    out[4+i] = add3(add3(scl3(va, r.x), scl3(vb, r.y)), add3(scl3(vc, r.z), aCa));
  }
}

// ---------------------------------------------------------------------------
// Kernel 1: 16x16 distance-matrix tiles via V_WMMA_F32_16X16X4_F32.
// dd_ij = |xi|^2 + |xj|^2 - 2 xi.xj ; Gram term is a K=3 (pad to 4) f32 WMMA.
// ---------------------------------------------------------------------------
__global__ void __launch_bounds__(256) dist_tiles_wmma(const float* __restrict__ xca,
                                                       const float* __restrict__ mask,
                                                       float* __restrict__ Dbuf){
  int wave = blockIdx.x * (blockDim.x >> 5) + (threadIdx.x >> 5);
  int tj = wave & 63;
  int ti = (wave >> 6) & 63;
  int b  = wave >> 12;                 // grid sized exactly: b in [0,8)
  int lane = threadIdx.x & 31;
  int r  = lane & 15;
  int hi = lane >> 4;

  int i = ti*16 + r;
  int j = tj*16 + r;
  const float* xi = xca + ((size_t)b*NN + i)*3;
  const float* xj = xca + ((size_t)b*NN + j)*3;
  float xix = xi[0], xiy = xi[1], xiz = xi[2];
  float xjx = xj[0], xjy = xj[1], xjz = xj[2];

  v2f afrag, bfrag;
  afrag.x = hi ? xiz : xix;  afrag.y = hi ? 0.0f : xiy;
  bfrag.x = hi ? xjz : xjx;  bfrag.y = hi ? 0.0f : xjy;

  v8f c = {};
  c = __builtin_amdgcn_wmma_f32_16x16x4_f32(false, afrag, false, bfrag,
                                            (short)0, c, false, false);

  float ni_l = xix*xix + xiy*xiy + xiz*xiz;     // lane l holds |row (l&15)|^2
  float nj_l = xjx*xjx + xjy*xjy + xjz*xjz;     // col N == own r
  float mi_l = mask[(size_t)b*NN + i];
  float mj   = mask[(size_t)b*NN + j];

  float* dbase = Dbuf + (size_t)b*NN*NN;
  #pragma unroll
  for (int v=0; v<8; ++v){
    int M = v + 8*hi;
    float niM = __shfl(ni_l, M, 32);
    float miM = __shfl(mi_l, M, 32);
    float dd  = fmaxf(niM + nj_l - 2.0f*c[v], 0.0f);
    float m2  = miM * mj;
    float D   = (1.0f - m2)*10000.0f + m2*sqrtf(dd + 1e-6f);
    dbase[(size_t)(ti*16 + M)*NN + (tj*16 + r)] = D;
  }
}

// ---------------------------------------------------------------------------
// Kernel 2: per-row top-k (k=30 smallest D; ties -> lower index, matching
// lax.top_k on -D). One 128-thread block per row, row cached in LDS.
// ---------------------------------------------------------------------------
__global__ void __launch_bounds__(128) topk_kernel(const float* __restrict__ Dbuf,
                                                   float* __restrict__ eidx_f){
  __shared__ float vals[NN];
  __shared__ float rv[128];
  __shared__ int   ri[128];
  int row = blockIdx.x;                        // b*NN + n
  const float* drow = Dbuf + (size_t)row*NN;
  for (int jj = threadIdx.x; jj < NN; jj += 128) vals[jj] = drow[jj];
  __syncthreads();
  for (int k = 0; k < KK; ++k){
    float bv = 3.4e38f; int bi = NN;
    for (int jj = threadIdx.x; jj < NN; jj += 128){
      float v = vals[jj];
      if (v < bv || (v == bv && jj < bi)) { bv = v; bi = jj; }
    }
    rv[threadIdx.x] = bv; ri[threadIdx.x] = bi;
    __syncthreads();
    for (int s = 64; s > 0; s >>= 1){
      if (threadIdx.x < s){
        float ov = rv[threadIdx.x + s]; int oi = ri[threadIdx.x + s];
        if (ov < rv[threadIdx.x] || (ov == rv[threadIdx.x] && oi < ri[threadIdx.x])){
          rv[threadIdx.x] = ov; ri[threadIdx.x] = oi;
        }
      }
      __syncthreads();
    }
    if (threadIdx.x == 0){
      eidx_f[(size_t)row*KK + k] = (float)ri[0];
      vals[ri[0]] = 3.4e38f;
    }
    __syncthreads();
  }
}

// ---------------------------------------------------------------------------
// Kernel 3: node features, one WAVE per residue; coalesced channel stores.
// _V row = [V_dist(192) | V_angles(12) | V_direct(9)]
// ---------------------------------------------------------------------------
__global__ void __launch_bounds__(256) node_kernel(const float* __restrict__ X,
                                                   const float* __restrict__ vraw,
                                                   float* __restrict__ Vout){
  __shared__ float sA[8][7][3];                // per-wave atom stage
  int w    = threadIdx.x >> 5;
  int lane = threadIdx.x & 31;
  int id   = blockIdx.x*8 + w;                 // node index, grid exact (8192)
  int b = id / NN, n = id % NN;
  float* vo = Vout + (size_t)id * VFEAT;

  if (lane == 0){
    F3 at[7];
    buildAtoms(X, vraw, b, n, at);
    #pragma unroll
    for (int q=0; q<7; ++q){ sA[w][q][0]=at[q].x; sA[w][q][1]=at[q].y; sA[w][q][2]=at[q].z; }
  }
  __syncthreads();

  // V_dist: 192 channels, lane-strided -> contiguous 128B stores
  for (int c = lane; c < 12*NRBF; c += 32){
    int p = c >> 4, t = c & 15;
    int i1 = c_NPa[p], i2 = c_NPb[p];
    F3 A = mk3(sA[w][i1][0], sA[w][i1][1], sA[w][i1][2]);
    F3 Bv= mk3(sA[w][i2][0], sA[w][i2][1], sA[w][i2][2]);
    F3 d = sub3(A, Bv);
    vo[c] = rbfv(sqrtf(dot3(d,d) + 1e-6f), t);
  }

  // Tail (21 values): lanes 0..11 -> angles, lanes 12..20 -> V_direct
  float tail = 0.0f;
  {
    // dihedral/bond angle for t = lane%3, padded index i = 3n+t-1
    int t = lane % 3;
    int i = 3*n + t - 1;
    float cD=1.f, sD=0.f, cA=1.f, sAn=0.f;
    if (i >= 0 && i <= 3*NN - 4){
      F3 u2 = chainU(X,b,i), u1 = chainU(X,b,i+1), u0 = chainU(X,b,i+2);
      F3 n2 = normeps(cross3(u2,u1));
      F3 n1 = normeps(cross3(u1,u0));
      float Dang = signf(dot3(u2,n1)) * acosf(clampc(dot3(n2,n1)));
      cD = cosf(Dang); sD = sinf(Dang);
      float A = acosf(clampc(dot3(u2,u1)));
      cA = cosf(A); sAn = sinf(A);
    }
    int g = lane / 3;                          // 0:cosD 1:sinD 2:cosA 3:sinA
    float av = (g==0) ? cD : (g==1) ? sD : (g==2) ? cA : sAn;

    // V_direct component for idx = lane-12: m = idx/3, a = idx%3
    F3 Qm[3];
    computeQ(X, b, n, Qm);
    int idx = (lane >= 12 && lane < 21) ? (lane - 12) : 0;
    int m = idx / 3, a = idx % 3;
    int ai = (m==0) ? 0 : (m==1) ? 2 : 3;      // atoms N, C, O
    F3 Xn = mk3(sA[w][0][0], sA[w][0][1], sA[w][0][2]);
    F3 ot = mk3(sA[w][ai][0], sA[w][ai][1], sA[w][ai][2]);
    F3 d  = sub3(ot, Xn);
    F3 pr = normeps(mk3(dot3(Qm[0],d), dot3(Qm[1],d), dot3(Qm[2],d)));
    float dv = comp3(pr, a);

    tail = (lane < 12) ? av : dv;
  }
  if (lane < 21) vo[192 + lane] = tail;
}

// ---------------------------------------------------------------------------
// Kernel 4: edge features, one WAVE per edge; coalesced channel stores.
// _E row = [E_dist(400) | quaternion(4) | E_direct(12)]
// ---------------------------------------------------------------------------
__global__ void __launch_bounds__(256) edge_kernel(const float* __restrict__ X,
                                                   const float* __restrict__ vraw,
                                                   const float* __restrict__ eidx_f,
                                                   float* __restrict__ Eout){
  __shared__ float sN[8][7][3];                // atoms of center n
  __shared__ float sJ[8][7][3];                // atoms of neighbor j
  int w    = threadIdx.x >> 5;
  int lane = threadIdx.x & 31;
  int e    = blockIdx.x*8 + w;                 // edge index, grid exact (245760)
  int b    = e / (NN*KK);
  int rem  = e % (NN*KK);
  int n    = rem / KK;
  int j    = (int)(eidx_f[e] + 0.5f);
  float* eo = Eout + (size_t)e * EFEAT;

  if (lane == 0){
    F3 at[7];
    buildAtoms(X, vraw, b, n, at);
    #pragma unroll
    for (int q=0; q<7; ++q){ sN[w][q][0]=at[q].x; sN[w][q][1]=at[q].y; sN[w][q][2]=at[q].z; }
    buildAtoms(X, vraw, b, j, at);
    #pragma unroll
    for (int q=0; q<7; ++q){ sJ[w][q][0]=at[q].x; sJ[w][q][1]=at[q].y; sJ[w][q][2]=at[q].z; }
  }
  __syncthreads();

  // E_dist: 400 channels, lane-strided -> contiguous 128B stores
  for (int c = lane; c < 25*NRBF; c += 32){
    int p = c >> 4, t = c & 15;
    int i1 = c_EPa[p], i2 = c_EPb[p];
    F3 A = mk3(sN[w][i1][0], sN[w][i1][1], sN[w][i1][2]);
    F3 Bv= mk3(sJ[w][i2][0], sJ[w][i2][1], sJ[w][i2][2]);
    F3 d = sub3(A, Bv);
    eo[c] = rbfv(sqrtf(dot3(d,d) + 1e-6f), t);
  }

  // Tail (16 values): lanes 0..3 -> quaternion, lanes 4..15 -> E_direct
  F3 Qm[3], Qj[3];
  computeQ(X, b, n, Qm);
  computeQ(X, b, j, Qj);

  // quaternion of Rrel = Qm^T * Qj  (redundant per lane; pure VALU)
  float R[3][3];
  #pragma unroll
  for (int a=0; a<3; ++a)
    #pragma unroll
    for (int cc=0; cc<3; ++cc)
      R[a][cc] = comp3(Qm[0],a)*comp3(Qj[0],cc)
               + comp3(Qm[1],a)*comp3(Qj[1],cc)
               + comp3(Qm[2],a)*comp3(Qj[2],cc);
  float Rxx=R[0][0], Ryy=R[1][1], Rzz=R[2][2];
  float m0 = 0.5f*sqrtf(fabsf(1.0f + Rxx - Ryy - Rzz));
  float m1 = 0.5f*sqrtf(fabsf(1.0f - Rxx + Ryy - Rzz));
  float m2 = 0.5f*sqrtf(fabsf(1.0f - Rxx - Ryy + Rzz));
  float qx = signf(R[2][1]-R[1][2]) * m0;
  float qy = signf(R[0][2]-R[2][0]) * m1;
  float qz = signf(R[1][0]-R[0][1]) * m2;
  float qw = 0.5f*sqrtf(fmaxf(1.0f + Rxx + Ryy + Rzz, 0.0f));
  float qn = fmaxf(sqrtf(qx*qx+qy*qy+qz*qz+qw*qw), 1e-12f);
  float qv = ((lane==0) ? qx : (lane==1) ? qy : (lane==2) ? qz : qw) / qn;

  // E_direct component for idx = lane-4: m = idx/3, a = idx%3; nb order (Ca,N,C,O)
  int idx = (lane >= 4 && lane < 16) ? (lane - 4) : 0;
  int m = idx / 3, a = idx % 3;
  int ai = (m==0) ? 1 : (m==1) ? 0 : m;        // {1,0,2,3}
  F3 Xn = mk3(sN[w][0][0], sN[w][0][1], sN[w][0][2]);
  F3 ot = mk3(sJ[w][ai][0], sJ[w][ai][1], sJ[w][ai][2]);
  F3 d  = sub3(ot, Xn);
  F3 pr = normeps(mk3(dot3(Qm[0],d), dot3(Qm[1],d), dot3(Qm[2],d)));
  float dv = comp3(pr, a);

  float tail = (lane < 4) ? qv : dv;
  if (lane < 16) eo[400 + lane] = tail;
}

extern "C" void kernel_launch(void* const* d_in, const int* in_sizes, int n_in,
                              void* d_out, int out_size, void* d_ws, size_t ws_size,
                              hipStream_t stream){
  (void)in_sizes; (void)n_in; (void)out_size; (void)d_ws; (void)ws_size;
  const float* X    = (const float*)d_in[1];   // (B,N,4,3)
  const float* mask = (const float*)d_in[2];   // (B,N)
  const float* xca  = (const float*)d_in[4];   // frame_t (B,N,3)
  const float* vraw = (const float*)d_in[10];  // virtual_atoms (3,3)

  float* out   = (float*)d_out;
  float* Vout  = out;                                        // 8192*213
  float* Eout  = out + (size_t)BB*NN*VFEAT;                  // 245760*416
  float* EidxF = Eout + (size_t)BB*NN*KK*EFEAT;              // 245760
  // Stage the B*N*N distance matrix (8.4M floats) inside the E region
  // (102M floats); it is consumed by topk before edge_kernel overwrites it.
  float* Dbuf  = Eout;

  // 1) distance tiles: 8 * 64 * 64 = 32768 waves, 8 waves/block
  dist_tiles_wmma<<<32768/8, 256, 0, stream>>>(xca, mask, Dbuf);
  // 2) top-k per row
  topk_kernel<<<BB*NN, 128, 0, stream>>>(Dbuf, EidxF);
  // 3) node features: 8192 waves, 8 per block
  node_kernel<<<BB*NN/8, 256, 0, stream>>>(X, vraw, Vout);
  // 4) edge features: 245760 waves, 8 per block (overwrites Dbuf with real E)
  edge_kernel<<<BB*NN*KK/8, 256, 0, stream>>>(X, vraw, EidxF, Eout);
}